// Time_Series_Tokenization_32779190403681
// MI455X (gfx1250) — compile-verified
//
#include <hip/hip_runtime.h>

typedef __attribute__((ext_vector_type(2))) float v2f;
typedef __attribute__((ext_vector_type(4))) float v4f;
typedef __attribute__((ext_vector_type(8))) float v8f;

#define B_    16
#define C_    64
#define L_    512
#define LP_   16
#define DM_   1024
#define NP_   64
#define P_    256      // period = L/2
#define HEAD_ 128      // p/2 leading NaN count
#define NV_   256      // valid trend samples = L - p
#define K_    48       // 3*L_P feature width
#define M_    (B_*C_)  // 1024 rows

// ---------------------------------------------------------------------------
// Kernel 1: per-(b,l) cross-channel stats -> m_norm[b,l] = mean / sqrt(var)
// 8192 outputs, each a 64-element reduction. Coalesced across l.
// ---------------------------------------------------------------------------
__global__ void stats_kernel(const float* __restrict__ x,
                             float* __restrict__ mnorm) {
    int t = blockIdx.x * blockDim.x + threadIdx.x;   // 0..B*L-1
    int b = t >> 9;                                  // / 512
    int l = t & (L_ - 1);
    const float* p = x + (size_t)b * C_ * L_ + l;
    float s = 0.f, ss = 0.f;
#pragma unroll 8
    for (int c = 0; c < C_; ++c) {
        float v = p[(size_t)c * L_];
        s  += v;
        ss += v * v;
    }
    float mean = s * (1.0f / C_);
    float var  = (ss - (float)C_ * mean * mean) * (1.0f / (C_ - 1));
    mnorm[t] = mean * rsqrtf(var);
}

// ---------------------------------------------------------------------------
// Kernel 2: per-series decomposition + three [512]->[16] projections.
// One 256-thread block per (b,c) series. Produces feats[series][48]
// in concat order [trend(16) | seasonal(16) | resid(16)].
//   tv[i]   = (0.5*x[i] + sum x[i+1..i+255] + 0.5*x[i+256]) / 256
//   dv[i]   = x[128+i] - tv[i],  m = mean(dv)
//   seas[l] = dv[(l+128) & 255] - m        (full length 512)
//   resid   = m on [128,384), 0 elsewhere  (constant — algebraic identity)
// ---------------------------------------------------------------------------
__global__ void __launch_bounds__(256)
decompose_kernel(const float* __restrict__ x,
                 const float* __restrict__ gamma,
                 const float* __restrict__ beta,
                 const float* __restrict__ wt, const float* __restrict__ bt,
                 const float* __restrict__ wsn, const float* __restrict__ bs,
                 const float* __restrict__ wr, const float* __restrict__ br,
                 const float* __restrict__ mnorm,
                 float* __restrict__ feats) {
    __shared__ float xs[L_];
    __shared__ float tv[NV_];
    __shared__ float seas[L_];
    __shared__ float red[8];
    __shared__ float mbar;

    int tid    = threadIdx.x;
    int lane   = tid & 31;
    int wave   = tid >> 5;
    int series = blockIdx.x;          // b*C + c
    int b      = series >> 6;         // / 64
    size_t base = (size_t)series * L_;

    // normalized input into LDS
    for (int l = tid; l < L_; l += 256) {
        float v = x[base + l];
        xs[l] = gamma[base + l] * (v - mnorm[b * L_ + l]) + beta[base + l];
    }
    __syncthreads();

    // centered moving average (valid region)
    if (tid < NV_) {
        float acc = 0.5f * (xs[tid] + xs[tid + P_]);
        for (int k = 1; k < P_; ++k) acc += xs[tid + k];
        tv[tid] = acc * (1.0f / P_);
    }
    __syncthreads();

    // dv and its mean m (block reduction over 256 values)
    float dvv = (tid < NV_) ? (xs[HEAD_ + tid] - tv[tid]) : 0.f;
    float a = dvv;
    for (int off = 16; off; off >>= 1) a += __shfl_xor(a, off, 32);
    if (lane == 0) red[wave] = a;
    __syncthreads();
    if (tid == 0) {
        float s = 0.f;
        for (int w = 0; w < 8; ++w) s += red[w];
        mbar = s * (1.0f / NV_);
    }
    __syncthreads();
    float m = mbar;

    // seasonal, full length
    for (int l = tid; l < L_; l += 256) {
        int i = (l + HEAD_) & (P_ - 1);
        seas[l] = (xs[HEAD_ + i] - tv[i]) - m;
    }
    __syncthreads();

    // 48 outputs: wave w handles j = 6w .. 6w+5; lanes stride the dot.
    for (int jj = 0; jj < 6; ++jj) {
        int j = wave * 6 + jj;
        float acc = 0.f;
        if (j < 16) {                           // trend projection
            const float* wrow = wt + (size_t)j * L_ + HEAD_;
            for (int i = lane; i < NV_; i += 32) acc += tv[i] * wrow[i];
        } else if (j < 32) {                    // seasonal projection
            const float* wrow = wsn + (size_t)(j - 16) * L_;
            for (int l = lane; l < L_; l += 32) acc += seas[l] * wrow[l];
        } else {                                // resid: m * sum(w over valid)
            const float* wrow = wr + (size_t)(j - 32) * L_ + HEAD_;
            for (int i = lane; i < NV_; i += 32) acc += wrow[i];
        }
        for (int off = 16; off; off >>= 1) acc += __shfl_xor(acc, off, 32);
        if (lane == 0) {
            float v;
            if (j < 16)      v = acc + bt[j];
            else if (j < 32) v = acc + bs[j - 16];
            else             v = m * acc + br[j - 32];
            feats[(size_t)series * K_ + j] = v;
        }
    }
}

// ---------------------------------------------------------------------------
// Kernel 3: out[m, n, d] = feats[m,:] @ w_g[d,:] + b_g[d], replicated over
// n = 0..63 patches.  M=1024, N=1024, K=48 via V_WMMA_F32_16X16X4_F32
// (12 K-steps). Each wave owns one 16x16 tile, stages it through LDS to get
// per-lane contiguous float4s, then streams it 64x with non-temporal b128
// stores (output = 256 MB > L2, so NT; this kernel is pure store-BW bound).
// ---------------------------------------------------------------------------
__global__ void __launch_bounds__(256)
gemm_bcast_kernel(const float* __restrict__ feats,
                  const float* __restrict__ wg,
                  const float* __restrict__ bg,
                  float* __restrict__ out) {
    __shared__ __align__(16) float tile[8][16][16];

    int tid  = threadIdx.x;
    int lane = tid & 31;
    int wave = tid >> 5;
    int t    = blockIdx.x * 8 + wave;   // tile id, 0..4095
    int nt   = t & 63;                  // N (d) tile
    int mt   = t >> 6;                  // M (row) tile
    int half = lane >> 4;               // selects K pair within A/B frags
    int l16  = lane & 15;

    int mrow = mt * 16 + l16;           // A-matrix row for this lane
    int dcol = nt * 16 + l16;           // B-matrix column (d) for this lane

    // C/D layout: acc[r] = element (M = r + 8*half, N = l16); init with b_g[N]
    float c0 = bg[dcol];
    v8f acc;
#pragma unroll
    for (int r = 0; r < 8; ++r) acc[r] = c0;

    const float* arow = feats + (size_t)mrow * K_;
    const float* brow = wg    + (size_t)dcol * K_;

#pragma unroll
    for (int kk = 0; kk < 12; ++kk) {
        int kbase = kk * 4 + half * 2;  // A 16x4: lanes 0-15 -> K0/K1, 16-31 -> K2/K3
        v2f av = { arow[kbase], arow[kbase + 1] };
        v2f bv = { brow[kbase], brow[kbase + 1] };
        acc = __builtin_amdgcn_wmma_f32_16x16x4_f32(
            /*neg_a=*/false, av, /*neg_b=*/false, bv,
            /*c_mod=*/(short)0, acc, /*reuse_a=*/false, /*reuse_b=*/false);
    }

    // stage result tile to LDS (each wave reads back only its own tile)
#pragma unroll
    for (int r = 0; r < 8; ++r)
        tile[wave][r + 8 * half][l16] = acc[r];
    __syncthreads();

    // lane -> (row = lane/2, 8 consecutive cols): two aligned float4s
    int row = lane >> 1;
    int col = (lane & 1) * 8;
    v4f f0 = *(const v4f*)&tile[wave][row][col];
    v4f f1 = *(const v4f*)&tile[wave][row][col + 4];

    size_t obase = ((size_t)(mt * 16 + row) * NP_) * DM_ + (size_t)nt * 16 + col;
    float* o = out + obase;
#pragma unroll 4
    for (int n = 0; n < NP_; ++n) {     // broadcast over patches
        __builtin_nontemporal_store(f0, (v4f*)o);
        __builtin_nontemporal_store(f1, (v4f*)(o + 4));
        o += DM_;
    }
}

// ---------------------------------------------------------------------------
extern "C" void kernel_launch(void* const* d_in, const int* in_sizes, int n_in,
                              void* d_out, int out_size, void* d_ws, size_t ws_size,
                              hipStream_t stream) {
    (void)in_sizes; (void)n_in; (void)out_size; (void)ws_size;
    const float* data_x = (const float*)d_in[0];
    const float* gamma  = (const float*)d_in[1];
    const float* beta   = (const float*)d_in[2];
    const float* w_t    = (const float*)d_in[3];
    const float* b_t    = (const float*)d_in[4];
    const float* w_s    = (const float*)d_in[5];
    const float* b_s    = (const float*)d_in[6];
    const float* w_r    = (const float*)d_in[7];
    const float* b_r    = (const float*)d_in[8];
    const float* w_g    = (const float*)d_in[9];
    const float* b_g    = (const float*)d_in[10];
    float* out = (float*)d_out;

    float* mnorm = (float*)d_ws;              // B*L   = 8192 floats (32 KB)
    float* feats = mnorm + B_ * L_;           // M*48  = 49152 floats (192 KB)

    stats_kernel<<<(B_ * L_) / 256, 256, 0, stream>>>(data_x, mnorm);
    decompose_kernel<<<M_, 256, 0, stream>>>(data_x, gamma, beta,
                                             w_t, b_t, w_s, b_s, w_r, b_r,
                                             mnorm, feats);
    gemm_bcast_kernel<<<(M_ / 16) * (DM_ / 16) / 8, 256, 0, stream>>>(feats, w_g, b_g, out);
}